// GCNRegressor_12446815224334
// MI455X (gfx1250) — compile-verified
//
#include <hip/hip_runtime.h>
#include <hip/hip_bf16.h>

// ---------------------------------------------------------------------------
// GCN regressor for MI455X (gfx1250, wave32, WMMA).
//   3 x [ t = h @ W (fp32 WMMA)  ->  agg = scatter_add(t[src]*ew, dst)
//         -> h = relu(agg + b) ]
//   pooled = segment_mean(h, batch) ; out = MLP head(pooled)
// ---------------------------------------------------------------------------

typedef __attribute__((ext_vector_type(2))) float v2f;
typedef __attribute__((ext_vector_type(8))) float v8f;

#define CH 64   // IN_CH == HID == 64
#define NG 64   // graphs

// ---------------------------------------------------------------------------
// t[N,64] = X[N,64] @ W[64,64]  via V_WMMA_F32_16X16X4_F32.
// Block = 128 threads = 4 waves; wave w computes the 16x16 tile at
// rows [16*blockIdx, +16), cols [16*w, +16). K=64 consumed in 16 steps of 4.
// N must be a multiple of 16 (50000 = 3125*16). No divergence: EXEC all ones.
// ---------------------------------------------------------------------------
__global__ __launch_bounds__(128) void gemm64_wmma(
    const float* __restrict__ X, const float* __restrict__ W,
    float* __restrict__ T) {
  const int lane = threadIdx.x & 31;
  const int wave = threadIdx.x >> 5;       // 0..3 -> output column tile
  const int row0 = blockIdx.x << 4;
  const int col0 = wave << 4;
  const int m    = lane & 15;
  const int kh   = (lane >> 4) << 1;       // lanes 0-15: K+{0,1}; 16-31: K+{2,3}

  v8f acc = {0.f, 0.f, 0.f, 0.f, 0.f, 0.f, 0.f, 0.f};

#pragma unroll
  for (int kk = 0; kk < 16; ++kk) {
    const int kb = (kk << 2) + kh;
    // A fragment (16x4): lane holds X[row0+m][kb], X[row0+m][kb+1] (contiguous)
    const float2 av = *(const float2*)(X + (size_t)(row0 + m) * CH + kb);
    v2f a; a.x = av.x; a.y = av.y;
    // B fragment (4x16): lane holds W[kb][col0+m], W[kb+1][col0+m]
    v2f b;
    b.x = W[(size_t)(kb + 0) * CH + col0 + m];
    b.y = W[(size_t)(kb + 1) * CH + col0 + m];
    acc = __builtin_amdgcn_wmma_f32_16x16x4_f32(
        /*neg_a=*/false, a, /*neg_b=*/false, b,
        /*c_mod=*/(short)0, acc, /*reuse_a=*/false, /*reuse_b=*/false);
  }

  // C/D layout: VGPR r -> row = row0 + r + 8*(lane>=16), col = col0 + (lane&15)
  const int rbase = row0 + ((lane >> 4) << 3);
#pragma unroll
  for (int r = 0; r < 8; ++r)
    T[(size_t)(rbase + r) * CH + col0 + m] = acc[r];
}

// ---------------------------------------------------------------------------
// agg[dst[e]][:] += t[src[e]][:] * ew[e]
// 16 threads per edge, one float4 (4 channels) per thread: the 16-thread
// gather is one contiguous 256B segment; 4 f32 atomics scatter per thread.
// ---------------------------------------------------------------------------
__global__ __launch_bounds__(256) void edge_scatter(
    const float* __restrict__ t, const int* __restrict__ src,
    const int* __restrict__ dst, const float* __restrict__ ew,
    float* __restrict__ agg, int E) {
  const int gid = blockIdx.x * blockDim.x + threadIdx.x;
  const int e   = gid >> 4;
  const int q   = gid & 15;                 // which float4 of the 64-ch row
  if (e >= E) return;

  // Look-ahead prefetch of a future source row (gfx1250 global_prefetch_b8).
  const int ep = e + 256;
  if (ep < E) __builtin_prefetch(t + (size_t)src[ep] * CH, 0, 0);

  const int   s = src[e];
  const int   d = dst[e];
  const float w = ew[e];

  const float4 v = ((const float4*)(t + (size_t)s * CH))[q];
  float* ap = agg + (size_t)d * CH + (q << 2);
  atomicAdd(ap + 0, v.x * w);
  atomicAdd(ap + 1, v.y * w);
  atomicAdd(ap + 2, v.z * w);
  atomicAdd(ap + 3, v.w * w);
}

// h = relu(agg + b)
__global__ __launch_bounds__(256) void bias_relu(
    const float* __restrict__ agg, const float* __restrict__ b,
    float* __restrict__ h, int total) {
  const int i = blockIdx.x * blockDim.x + threadIdx.x;
  if (i >= total) return;
  const float v = agg[i] + b[i & (CH - 1)];
  h[i] = v > 0.f ? v : 0.f;
}

// sums[batch[node]][c] += h[node][c]
__global__ __launch_bounds__(256) void pool_accum(
    const float* __restrict__ h, const int* __restrict__ batch,
    float* __restrict__ sums, int N) {
  const int i = blockIdx.x * blockDim.x + threadIdx.x;
  if (i >= N * CH) return;
  const int node = i >> 6;
  const int c    = i & (CH - 1);
  atomicAdd(&sums[(size_t)batch[node] * CH + c], h[i]);
}

__global__ __launch_bounds__(256) void pool_count(
    const int* __restrict__ batch, float* __restrict__ cnts, int N) {
  const int i = blockIdx.x * blockDim.x + threadIdx.x;
  if (i < N) atomicAdd(&cnts[batch[i]], 1.0f);
}

// MLP head: one block of 64 threads, one graph per thread.
// pooled[64] -> relu(@lw1[64,32]+lb1) -> relu(@lw2[32,8]+lb2) -> @lw3[8,1]+lb3
__global__ __launch_bounds__(64) void head_mlp(
    const float* __restrict__ sums, const float* __restrict__ cnts,
    const float* __restrict__ lw1, const float* __restrict__ lb1,
    const float* __restrict__ lw2, const float* __restrict__ lb2,
    const float* __restrict__ lw3, const float* __restrict__ lb3,
    float* __restrict__ out) {
  const int g = threadIdx.x;
  if (g >= NG) return;
  float c = cnts[g];
  c = c < 1.f ? 1.f : c;
  const float inv = 1.f / c;

  float p[CH];
#pragma unroll
  for (int k = 0; k < CH; ++k) p[k] = sums[(size_t)g * CH + k] * inv;

  float z1[32];
  for (int j = 0; j < 32; ++j) {
    float s = lb1[j];
    for (int k = 0; k < CH; ++k) s += p[k] * lw1[k * 32 + j];
    z1[j] = s > 0.f ? s : 0.f;
  }
  float z2[8];
  for (int j = 0; j < 8; ++j) {
    float s = lb2[j];
    for (int k = 0; k < 32; ++k) s += z1[k] * lw2[k * 8 + j];
    z2[j] = s > 0.f ? s : 0.f;
  }
  float s = lb3[0];
  for (int k = 0; k < 8; ++k) s += z2[k] * lw3[k];
  out[g] = s;
}

// ---------------------------------------------------------------------------
extern "C" void kernel_launch(void* const* d_in, const int* in_sizes, int n_in,
                              void* d_out, int out_size, void* d_ws, size_t ws_size,
                              hipStream_t stream) {
  const float* x     = (const float*)d_in[0];
  const int*   eidx  = (const int*)d_in[1];
  const float* ew    = (const float*)d_in[2];
  const int*   batch = (const int*)d_in[3];
  const float* W[3]  = {(const float*)d_in[4], (const float*)d_in[6],
                        (const float*)d_in[8]};
  const float* B[3]  = {(const float*)d_in[5], (const float*)d_in[7],
                        (const float*)d_in[9]};
  const float* lw1 = (const float*)d_in[10];
  const float* lb1 = (const float*)d_in[11];
  const float* lw2 = (const float*)d_in[12];
  const float* lb2 = (const float*)d_in[13];
  const float* lw3 = (const float*)d_in[14];
  const float* lb3 = (const float*)d_in[15];

  const int N = in_sizes[0] / CH;   // 50000
  const int E = in_sizes[2];        // 800000
  const int* src = eidx;
  const int* dst = eidx + E;

  // Workspace layout (fp32 elements)
  float* t    = (float*)d_ws;            // [N,64] GEMM output
  float* agg  = t    + (size_t)N * CH;   // [N,64] scatter accumulator
  float* hA   = agg  + (size_t)N * CH;   // [N,64] ping
  float* hB   = hA   + (size_t)N * CH;   // [N,64] pong
  float* sums = hB   + (size_t)N * CH;   // [64,64]
  float* cnts = sums + NG * CH;          // [64]   (contiguous after sums)

  const int gemm_blocks = N / 16;                       // 3125
  const int edge_blocks = (E * 16 + 255) / 256;         // 50000
  const int elem_blocks = (N * CH + 255) / 256;         // 12500

  const float* in = x;
  float* outbuf = hA;
  for (int l = 0; l < 3; ++l) {
    gemm64_wmma<<<gemm_blocks, 128, 0, stream>>>(in, W[l], t);
    hipMemsetAsync(agg, 0, (size_t)N * CH * sizeof(float), stream);
    edge_scatter<<<edge_blocks, 256, 0, stream>>>(t, src, dst, ew, agg, E);
    bias_relu<<<elem_blocks, 256, 0, stream>>>(agg, B[l], outbuf, N * CH);
    in = outbuf;
    outbuf = (outbuf == hA) ? hB : hA;
  }

  hipMemsetAsync(sums, 0, (size_t)(NG * CH + NG) * sizeof(float), stream);
  pool_accum<<<elem_blocks, 256, 0, stream>>>(in, batch, sums, N);
  pool_count<<<(N + 255) / 256, 256, 0, stream>>>(batch, cnts, N);
  head_mlp<<<1, 64, 0, stream>>>(sums, cnts, lw1, lb1, lw2, lb2, lw3, lb3,
                                 (float*)d_out);
}